// H2FDMultiRelationLayer_72748156059659
// MI455X (gfx1250) — compile-verified
//
#include <hip/hip_runtime.h>
#include <hip/hip_bf16.h>
#include <math.h>

// Problem constants (from reference)
#define NN   50000   // nodes
#define EE   300000  // edges per relation
#define INF_ 128     // in_feats
#define HH   256     // hidden width H = HF*AH
#define AHH  4       // attention heads
#define HFF  64      // h_feats per head
#define RR   3       // relations

typedef __attribute__((ext_vector_type(2))) float v2f;
typedef __attribute__((ext_vector_type(4))) float v4f;
typedef __attribute__((ext_vector_type(8))) float v8f;

// ---------------- float atomic max via int/uint views ----------------
__device__ __forceinline__ void atomicMaxF(float* addr, float val) {
    if (val >= 0.0f) atomicMax((int*)addr, __float_as_int(val));
    else             atomicMin((unsigned int*)addr, __float_as_uint(val));
}

// ---------------- WMMA f32 GEMM (NF = 256 fixed, K compile-time) ----------------
// One wave computes one 16x16 tile of C with V_WMMA_F32_16X16X4_F32, K/4 steps,
// fully unrolled so every load has an immediate offset (no per-step address math).
// MODE==0: C = A@B + bias[n];  MODE==1: C += A@B.
// A: [M,K] row-major, B: [K,256] row-major.  Grids are exact so EXEC is all-ones.
template <int MODE, int K>
__global__ void gemm_wmma_f32(const float* __restrict__ A,
                              const float* __restrict__ B,
                              const float* __restrict__ bias,
                              float* __restrict__ C) {
    constexpr int NF = HH;                       // 256
    const int lane  = threadIdx.x & 31;
    const int wave  = threadIdx.x >> 5;
    const int wg    = blockIdx.x * 8 + wave;     // global tile index
    const int mt    = wg >> 4;                   // NF/16 == 16 tiles per row
    const int nt    = wg & 15;
    const int r16   = lane & 15;                 // A row / B,C column within tile
    const int khalf = (lane >> 4) << 1;          // lanes 0-15 -> K+0/1, 16-31 -> K+2/3
    const int mofs  = (lane >> 4) << 3;          // C rows: 0..7 or 8..15

    const float* Arow = A + (size_t)(mt * 16 + r16) * K + khalf;
    const float* Bcol = B + (size_t)khalf * NF + nt * 16 + r16;
    float*       Ccol = C + (size_t)(mt * 16 + mofs) * NF + nt * 16 + r16;

    v8f c;
    if (MODE == 0) {
        const float bv = bias[nt * 16 + r16];
        #pragma unroll
        for (int i = 0; i < 8; ++i) c[i] = bv;
    } else {
        #pragma unroll
        for (int i = 0; i < 8; ++i) c[i] = Ccol[(size_t)i * NF];
    }

    #pragma unroll
    for (int k = 0; k < K; k += 4) {
        const v2f a = *(const v2f*)(Arow + k);   // 8B-aligned: khalf even, k%4==0
        v2f b;
        b.x = Bcol[(size_t)k * NF];
        b.y = Bcol[(size_t)(k + 1) * NF];
        c = __builtin_amdgcn_wmma_f32_16x16x4_f32(false, a, false, b,
                                                  (short)0, c, false, false);
    }

    #pragma unroll
    for (int i = 0; i < 8; ++i) Ccol[(size_t)i * NF] = c[i];
}

// ---------------- per-node scalars for sign: p,q ----------------
// p[n] = hd[n,:] . (fW[0:256] + fW[512:768])
// q[n] = hd[n,:] . (fW[256:512] - fW[512:768])
__global__ void node_pq_kernel(const float* __restrict__ hd,
                               const float* __restrict__ fW,
                               float* __restrict__ p, float* __restrict__ q) {
    const int lane = threadIdx.x & 31;
    const int n    = blockIdx.x * 8 + (threadIdx.x >> 5);
    const float* row = hd + (size_t)n * HH + lane * 8;
    const v4f x0 = *(const v4f*)row;
    const v4f x1 = *(const v4f*)(row + 4);
    const v4f f00 = *(const v4f*)(fW + lane * 8);
    const v4f f01 = *(const v4f*)(fW + lane * 8 + 4);
    const v4f f10 = *(const v4f*)(fW + 256 + lane * 8);
    const v4f f11 = *(const v4f*)(fW + 256 + lane * 8 + 4);
    const v4f f20 = *(const v4f*)(fW + 512 + lane * 8);
    const v4f f21 = *(const v4f*)(fW + 512 + lane * 8 + 4);
    float sp = 0.f, sq = 0.f;
    #pragma unroll
    for (int i = 0; i < 4; ++i) {
        sp += x0[i] * (f00[i] + f20[i]) + x1[i] * (f01[i] + f21[i]);
        sq += x0[i] * (f10[i] - f20[i]) + x1[i] * (f11[i] - f21[i]);
    }
    #pragma unroll
    for (int m = 16; m >= 1; m >>= 1) {
        sp += __shfl_xor(sp, m, 32);
        sq += __shfl_xor(sq, m, 32);
    }
    if (lane == 0) { p[n] = sp; q[n] = sq; }
}

// ---------------- per-node per-head scalars: u,v ----------------
// u[n,a] = hw[n, a*64 : a*64+64] . aWr[0:64]
// v[n,a] = hw[n, a*64 : a*64+64] . aWr[64:128]
__global__ void node_uv_kernel(const float* __restrict__ hw,
                               const float* __restrict__ aWr,
                               float* __restrict__ u, float* __restrict__ v) {
    const int lane = threadIdx.x & 31;
    const int n    = blockIdx.x * 8 + (threadIdx.x >> 5);
    const int a    = lane >> 3;          // 8 lanes per head
    const int f0   = (lane & 7) * 8;     // 8 contiguous features per lane
    const float* row = hw + (size_t)n * HH + a * HFF + f0;
    const v4f x0 = *(const v4f*)row;
    const v4f x1 = *(const v4f*)(row + 4);
    const v4f wa0 = *(const v4f*)(aWr + f0);
    const v4f wa1 = *(const v4f*)(aWr + f0 + 4);
    const v4f wb0 = *(const v4f*)(aWr + HFF + f0);
    const v4f wb1 = *(const v4f*)(aWr + HFF + f0 + 4);
    float su = 0.f, sv = 0.f;
    #pragma unroll
    for (int i = 0; i < 4; ++i) {
        su += x0[i] * wa0[i] + x1[i] * wa1[i];
        sv += x0[i] * wb0[i] + x1[i] * wb1[i];
    }
    su += __shfl_xor(su, 1, 32); su += __shfl_xor(su, 2, 32); su += __shfl_xor(su, 4, 32);
    sv += __shfl_xor(sv, 1, 32); sv += __shfl_xor(sv, 2, 32); sv += __shfl_xor(sv, 4, 32);
    if ((lane & 7) == 0) { u[n * AHH + a] = su; v[n * AHH + a] = sv; }
}

// ---------------- init kernels ----------------
__global__ void init_out_kernel(float* __restrict__ out, const float* __restrict__ linb) {
    const int idx = blockIdx.x * 256 + threadIdx.x;   // grid covers NN*HH exactly
    out[idx] = linb[idx & (HH - 1)];
}

__global__ void init_rel_kernel(float* __restrict__ amax, float* __restrict__ denom,
                                float* __restrict__ out_r) {
    const int idx = blockIdx.x * 256 + threadIdx.x;   // grid covers NN*HH exactly
    out_r[idx] = 0.f;
    if (idx < NN * AHH) {
        amax[idx]  = -INFINITY;
        denom[idx] = 0.f;
    }
}

// ---------------- edge pass A: sign, alpha, segment max ----------------
__global__ void edge_a_kernel(const int* __restrict__ src, const int* __restrict__ dst,
                              const float* __restrict__ p, const float* __restrict__ q,
                              const float* __restrict__ fb,
                              const float* __restrict__ u, const float* __restrict__ v,
                              const float* __restrict__ abr,
                              float* __restrict__ sign, float* __restrict__ alpha,
                              float* __restrict__ amax) {
    const int e = blockIdx.x * 256 + threadIdx.x;
    if (e >= EE) return;
    const int s = src[e], d = dst[e];
    const float sc = p[s] + q[d] + fb[0];
    const float sg = (sc > 0.f) ? 1.f : ((sc < 0.f) ? -1.f : 0.f);
    sign[e] = sg;
    const float ab0 = abr[0];
    #pragma unroll
    for (int a = 0; a < AHH; ++a) {
        float al = sg * u[s * AHH + a] + v[d * AHH + a] + ab0;
        al = (al > 0.f) ? al : 0.01f * al;           // leaky_relu 0.01
        alpha[(size_t)e * AHH + a] = al;
        atomicMaxF(&amax[d * AHH + a], al);
    }
}

// ---------------- edge pass B: exp and segment sum ----------------
__global__ void edge_b_kernel(const int* __restrict__ dst,
                              float* __restrict__ alpha,   // in: alpha, out: exp(alpha-max)
                              const float* __restrict__ amax,
                              float* __restrict__ denom) {
    const int e = blockIdx.x * 256 + threadIdx.x;
    if (e >= EE) return;
    const int d = dst[e];
    #pragma unroll
    for (int a = 0; a < AHH; ++a) {
        const float ex = __expf(alpha[(size_t)e * AHH + a] - amax[d * AHH + a]);
        alpha[(size_t)e * AHH + a] = ex;
        unsafeAtomicAdd(&denom[d * AHH + a], ex);
    }
}

// ---------------- edge pass C: weighted scatter-add ----------------
// One wave per edge.  Features are assigned per-lane-contiguously
// (idx = lane*8 + j), so head = lane>>3 is lane-constant: one softmax weight
// per lane, and the 8 gathered features become two global_load_b128.
__global__ void edge_c_kernel(const int* __restrict__ src, const int* __restrict__ dst,
                              const float* __restrict__ sign,
                              const float* __restrict__ ex,
                              const float* __restrict__ denom,
                              const float* __restrict__ hw,
                              float* __restrict__ out_r) {
    const int e = blockIdx.x * 8 + (threadIdx.x >> 5);   // grid covers EE exactly
    const int lane = threadIdx.x & 31;
    const int a = lane >> 3;                             // head for this lane
    const int s = src[e], d = dst[e];
    const float wgt = sign[e] * ex[(size_t)e * AHH + a] / denom[d * AHH + a];
    const float* hsrc = hw + (size_t)s * HH + lane * 8;
    float* od = out_r + (size_t)d * HH + lane * 8;
    const v4f x0 = *(const v4f*)hsrc;         // 32B-aligned
    const v4f x1 = *(const v4f*)(hsrc + 4);
    #pragma unroll
    for (int j = 0; j < 4; ++j) unsafeAtomicAdd(&od[j],     wgt * x0[j]);
    #pragma unroll
    for (int j = 0; j < 4; ++j) unsafeAtomicAdd(&od[4 + j], wgt * x1[j]);
}

// ---------------- host launcher ----------------
extern "C" void kernel_launch(void* const* d_in, const int* in_sizes, int n_in,
                              void* d_out, int out_size, void* d_ws, size_t ws_size,
                              hipStream_t stream) {
    (void)in_sizes; (void)n_in; (void)out_size; (void)ws_size;
    const float* h    = (const float*)d_in[0];
    const float* dW   = (const float*)d_in[1];
    const float* db   = (const float*)d_in[2];
    const float* fW   = (const float*)d_in[3];
    const float* fb   = (const float*)d_in[4];
    const float* wW   = (const float*)d_in[5];
    const float* wb   = (const float*)d_in[6];
    const float* aW   = (const float*)d_in[7];
    const float* ab   = (const float*)d_in[8];
    const float* linW = (const float*)d_in[9];
    const float* linb = (const float*)d_in[10];
    const int*   src  = (const int*)d_in[11];
    const int*   dst  = (const int*)d_in[12];
    float* out = (float*)d_out;

    // bump allocator over d_ws (total ~163 MB)
    char* wp = (char*)d_ws;
    auto alloc = [&](size_t bytes) -> float* {
        char* r = wp;
        wp += (bytes + 255) & ~(size_t)255;
        return (float*)r;
    };
    float* hd    = alloc((size_t)NN * HH * 4);   // 51.2 MB
    float* hw    = alloc((size_t)NN * HH * 4);   // 51.2 MB (reused per relation)
    float* out_r = alloc((size_t)NN * HH * 4);   // 51.2 MB (reused per relation)
    float* p     = alloc((size_t)NN * 4);
    float* q     = alloc((size_t)NN * 4);
    float* u     = alloc((size_t)NN * AHH * 4);
    float* v     = alloc((size_t)NN * AHH * 4);
    float* amax  = alloc((size_t)NN * AHH * 4);
    float* denom = alloc((size_t)NN * AHH * 4);
    float* sgn   = alloc((size_t)EE * 4);
    float* albuf = alloc((size_t)EE * AHH * 4);

    const int gemm_blocks  = (NN / 16) * (HH / 16) / 8;  // 6250, exact
    const int node_blocks  = NN / 8;                     // 6250, exact
    const int elem_blocks  = (NN * HH) / 256;            // 50000, exact
    const int edge_blocks  = (EE + 255) / 256;
    const int edgec_blocks = EE / 8;                     // 37500, exact

    // hd = h @ dW + db
    gemm_wmma_f32<0, INF_><<<gemm_blocks, 256, 0, stream>>>(h, dW, db, hd);
    // per-node sign components
    node_pq_kernel<<<node_blocks, 256, 0, stream>>>(hd, fW, p, q);
    // out = linb (broadcast); accumulated by per-relation GEMMs below
    init_out_kernel<<<elem_blocks, 256, 0, stream>>>(out, linb);

    for (int r = 0; r < RR; ++r) {
        // hw = h @ wW[r] + wb[r]
        gemm_wmma_f32<0, INF_><<<gemm_blocks, 256, 0, stream>>>(
            h, wW + (size_t)r * INF_ * HH, wb + (size_t)r * HH, hw);
        // per-node per-head attention components
        node_uv_kernel<<<node_blocks, 256, 0, stream>>>(hw, aW + (size_t)r * 2 * HFF, u, v);
        // reset segment-softmax state + relation output
        init_rel_kernel<<<elem_blocks, 256, 0, stream>>>(amax, denom, out_r);
        // edge passes
        edge_a_kernel<<<edge_blocks, 256, 0, stream>>>(
            src + (size_t)r * EE, dst + (size_t)r * EE, p, q, fb, u, v, ab + r,
            sgn, albuf, amax);
        edge_b_kernel<<<edge_blocks, 256, 0, stream>>>(
            dst + (size_t)r * EE, albuf, amax, denom);
        edge_c_kernel<<<edgec_blocks, 256, 0, stream>>>(
            src + (size_t)r * EE, dst + (size_t)r * EE, sgn, albuf, denom, hw, out_r);
        // out += out_r @ linW[r*256:(r+1)*256, :]
        gemm_wmma_f32<1, HH><<<gemm_blocks, 256, 0, stream>>>(
            out_r, linW + (size_t)r * HH * HH, nullptr, out);
    }
}